// BaseRuleLearner_10342281249199
// MI455X (gfx1250) — compile-verified
//
#include <hip/hip_runtime.h>

typedef float v2f __attribute__((ext_vector_type(2)));
typedef float v8f __attribute__((ext_vector_type(8)));

#define B_TOT   4096
#define O_OBJ   8
#define E_DIM   64
#define I_RULES 4
#define NPERM   336            // P(8,3)
#define WAVES_PER_BLOCK 4

#define SB 49                  // padded stride (float2) for 48 rule-binary cols
#define SU 17                  // padded stride (float2) for 16 rule-unary cols
#define MSTRIDE 65             // padded row stride (floats) for 36x64 match matrix
#define MATCH_FLOATS (36 * MSTRIDE)          // 2340
#define WAVE_REGION  (MATCH_FLOATS + 96 + 12) // + 12x8 unary match + pad = 2448

__global__ __launch_bounds__(128)
void BaseRuleLearner_kernel(const float* __restrict__ unary,
                            const float* __restrict__ binary,
                            const float* __restrict__ rule_u,
                            const float* __restrict__ rule_b,
                            float* __restrict__ out) {
    // B-operand staging in WMMA fragment layout: [(kt*2+hi)*stride + n] = (e0, e0+1)
    __shared__ __align__(16) float2 sBbin[16 * 2 * SB];   // 12544 B
    __shared__ __align__(16) float2 sBun [16 * 2 * SU];   //  4352 B
    __shared__ __align__(16) float  sMatch[WAVES_PER_BLOCK * WAVE_REGION]; // 39168 B

    const int tid = threadIdx.x;

    // ---- Stage rule_binary (36 rows x 64 e) into B-fragment layout ----
    const float4* rb4 = reinterpret_cast<const float4*>(rule_b);
    for (int t = tid; t < 576; t += 128) {           // 36*16 float4s
        const int n = t >> 4;                        // row i*9+n*3+m
        const int c = t & 15;                        // e/4 (= kt)
        float4 v = rb4[t];
        sBbin[(c * 2 + 0) * SB + n] = make_float2(v.x, v.y);
        sBbin[(c * 2 + 1) * SB + n] = make_float2(v.z, v.w);
    }
    for (int t = tid; t < 32 * 12; t += 128) {       // zero cols 36..47
        const int kh = t / 12;
        sBbin[kh * SB + 36 + (t % 12)] = make_float2(0.f, 0.f);
    }
    // ---- Stage rule_unary (12 rows x 64 e) ----
    const float4* ru4 = reinterpret_cast<const float4*>(rule_u);
    for (int t = tid; t < 192; t += 128) {           // 12*16 float4s
        const int n = t >> 4;                        // row i*3+k
        const int c = t & 15;
        float4 v = ru4[t];
        sBun[(c * 2 + 0) * SU + n] = make_float2(v.x, v.y);
        sBun[(c * 2 + 1) * SU + n] = make_float2(v.z, v.w);
    }
    for (int t = tid; t < 32 * 4; t += 128) {        // zero cols 12..15
        sBun[(t >> 2) * SU + 12 + (t & 3)] = make_float2(0.f, 0.f);
    }
    __syncthreads();

    const int wave = tid >> 5;
    const int lane = tid & 31;
    const int lm   = lane & 15;
    const int hi   = lane >> 4;
    const int b    = blockIdx.x * WAVES_PER_BLOCK + wave;

    const float2* Ab = reinterpret_cast<const float2*>(binary) + (size_t)b * (O_OBJ * O_OBJ * E_DIM / 2);
    const float2* Au = reinterpret_cast<const float2*>(unary)  + (size_t)b * (O_OBJ * E_DIM / 2);

    v8f accB[4][3];
    v8f accU = {};
#pragma unroll
    for (int mt = 0; mt < 4; ++mt)
#pragma unroll
        for (int nt = 0; nt < 3; ++nt) accB[mt][nt] = {};

    // ---- GEMM: D(64x48) = binary_feats[b](64x64) * rule_binaryT, K in steps of 4 ----
#pragma unroll
    for (int kt = 0; kt < 16; ++kt) {
        const int e2 = kt * 2 + hi;                  // float2 col index = (kt*4 + 2*hi)/2
        v2f aB[4], bB[3];
#pragma unroll
        for (int mt = 0; mt < 4; ++mt)               // A frag: lane=row, 2 consecutive e
            aB[mt] = *reinterpret_cast<const v2f*>(Ab + (mt * 16 + lm) * 32 + e2);
#pragma unroll
        for (int nt = 0; nt < 3; ++nt)
            bB[nt] = *reinterpret_cast<const v2f*>(&sBbin[e2 * SB + nt * 16 + lm]);
        v2f aU = *reinterpret_cast<const v2f*>(Au + (lane & 7) * 32 + e2); // rows 8..15 dup (unread)
        v2f bU = *reinterpret_cast<const v2f*>(&sBun[e2 * SU + lm]);

#pragma unroll
        for (int mt = 0; mt < 4; ++mt)
#pragma unroll
            for (int nt = 0; nt < 3; ++nt)
                accB[mt][nt] = __builtin_amdgcn_wmma_f32_16x16x4_f32(
                    false, aB[mt], false, bB[nt], (short)0, accB[mt][nt], false, false);
        accU = __builtin_amdgcn_wmma_f32_16x16x4_f32(
            false, aU, false, bU, (short)0, accU, false, false);
    }

    // ---- Spill match matrices to per-wave LDS region ----
    float* mW = sMatch + wave * WAVE_REGION;         // match[n][jk], stride 65
    float* uW = mW + MATCH_FLOATS;                   // umatch[i*3+k][obj], stride 8
#pragma unroll
    for (int nt = 0; nt < 3; ++nt) {
        const int n = nt * 16 + lm;
        if (n < 36) {
#pragma unroll
            for (int mt = 0; mt < 4; ++mt)
#pragma unroll
                for (int g = 0; g < 8; ++g)
                    mW[n * MSTRIDE + mt * 16 + hi * 8 + g] = accB[mt][nt][g];
        }
    }
    if (hi == 0 && lm < 12) {
#pragma unroll
        for (int g = 0; g < 8; ++g) uW[lm * 8 + g] = accU[g];
    }
    // same-wave DS ops are in-order: no barrier needed before gather

    // ---- Permutation scoring: min over 336 perms per rule ----
    float mins[I_RULES] = {1e30f, 1e30f, 1e30f, 1e30f};
    for (int p = lane; p < NPERM; p += 32) {
        const int a0  = p / 42;
        const int rem = p - a0 * 42;
        const int b0  = rem / 6;
        const int c0  = rem - b0 * 6;
        const int ob  = b0 + (b0 >= a0 ? 1 : 0);
        const int lo  = (a0 < ob) ? a0 : ob;
        const int hb  = (a0 < ob) ? ob : a0;
        int oc = c0 + (c0 >= lo ? 1 : 0);
        oc += (oc >= hb ? 1 : 0);
        const int obj[3] = {a0, ob, oc};
#pragma unroll
        for (int i = 0; i < I_RULES; ++i) {
            float s = uW[(i * 3 + 0) * 8 + obj[0]]
                    + uW[(i * 3 + 1) * 8 + obj[1]]
                    + uW[(i * 3 + 2) * 8 + obj[2]];
#pragma unroll
            for (int n = 0; n < 3; ++n)
#pragma unroll
                for (int m = 0; m < 3; ++m)
                    s += mW[(i * 9 + n * 3 + m) * MSTRIDE + obj[n] * 8 + obj[m]];
            mins[i] = fminf(mins[i], s);
        }
    }
#pragma unroll
    for (int off = 16; off > 0; off >>= 1)
#pragma unroll
        for (int i = 0; i < I_RULES; ++i)
            mins[i] = fminf(mins[i], __shfl_xor(mins[i], off, 32));

    if (lane == 0) {
        const float mx = fmaxf(fmaxf(mins[0], mins[1]), fmaxf(mins[2], mins[3]));
        const float e0 = __expf(mins[0] - mx), e1 = __expf(mins[1] - mx);
        const float e2 = __expf(mins[2] - mx), e3 = __expf(mins[3] - mx);
        const float inv = 1.0f / (e0 + e1 + e2 + e3);
        reinterpret_cast<float4*>(out)[b] =
            make_float4((e0 + e1) * inv, (e2 + e3) * inv, 0.f, 0.f);
    }
}

extern "C" void kernel_launch(void* const* d_in, const int* in_sizes, int n_in,
                              void* d_out, int out_size, void* d_ws, size_t ws_size,
                              hipStream_t stream) {
    const float* unary  = (const float*)d_in[0];  // (4096, 8, 64)
    const float* binary = (const float*)d_in[1];  // (4096, 8, 8, 64)
    const float* rule_u = (const float*)d_in[2];  // (4, 3, 64)
    const float* rule_b = (const float*)d_in[3];  // (4, 3, 3, 64)
    float* out = (float*)d_out;                   // (4096, 4)
    dim3 grid(B_TOT / WAVES_PER_BLOCK);           // 1024 blocks x 128 threads (4 waves)
    BaseRuleLearner_kernel<<<grid, 128, 0, stream>>>(unary, binary, rule_u, rule_b, out);
}